// Region_80788334838504
// MI455X (gfx1250) — compile-verified
//
#include <hip/hip_runtime.h>
#include <cstdint>
#include <cstddef>

// ---------------- problem constants ----------------
#define NNODE 10000
#define NEDGE 100000
#define NREG  100
#define NGRAPH 4
#define DIM   256           // L*M = 4*64
#define FNODE 100
#define RAW   356           // DIM + FNODE
#define EMB   128
#define MSGD  64
#define HID   178           // RAW/2
#define HIDP  180           // HID padded to multiple of 4
#define MAXDEG 16           // actual degree is exactly 10 by construction

__constant__ float c_LAMB[4] = {0.1f, 0.05f, 0.01f, 0.005f};

typedef __attribute__((ext_vector_type(2))) float v2f;
typedef __attribute__((ext_vector_type(8))) float v8f;

// ---------------- utility kernels ----------------
__global__ void k_zero32(unsigned int* p, long n) {
  long i = (long)blockIdx.x * blockDim.x + threadIdx.x;
  if (i < n) p[i] = 0u;
}

// Pass 1 over edges: t_last = segment_max(timestamps, src) via ordered-uint
// atomicMax (timestamps > 0), and bucket edge ids by source node.
__global__ void k_edge_scan(const int* __restrict__ src, const float* __restrict__ ts,
                            unsigned int* __restrict__ tlast_u, int* __restrict__ cnt,
                            int* __restrict__ bucket, int E) {
  int e = blockIdx.x * blockDim.x + threadIdx.x;
  if (e >= E) return;
  int s = src[e];
  atomicMax(&tlast_u[s], __float_as_uint(ts[e]));
  int slot = atomicAdd(&cnt[s], 1);
  if (slot < MAXDEG) bucket[s * MAXDEG + slot] = e;
}

// One block (384 threads) per node: gather its edges, accumulate
// agg[n,l,f] = sum_e w[e,l] * src_msg[e,f]  (f<356 -> core, f==356 -> cnt).
__global__ void k_node_agg(const float* __restrict__ memory, const float* __restrict__ nodef,
                           const float* __restrict__ ts, const float* __restrict__ td,
                           const int* __restrict__ tgt, const float* __restrict__ tlast,
                           const int* __restrict__ cnt, const int* __restrict__ bucket,
                           float* __restrict__ core, float* __restrict__ cnt4) {
  int n = blockIdx.x;
  int f = threadIdx.x;
  __shared__ float s_w[MAXDEG][4];    // w[e,l] = exp(-lamb_l*(t_last - ts))
  __shared__ float s_dec[MAXDEG][4];  // dec[e,l] = exp(-lamb_l*td)
  __shared__ int   s_tgt[MAXDEG];
  int deg = cnt[n]; if (deg > MAXDEG) deg = MAXDEG;
  float tl = tlast[n];
  if (f < deg) {
    int e = bucket[n * MAXDEG + f];
    s_tgt[f] = tgt[e];
    float dt = tl - ts[e];
    float d  = td[e];
    #pragma unroll
    for (int l = 0; l < 4; ++l) {
      s_w[f][l]   = expf(-c_LAMB[l] * dt);
      s_dec[f][l] = expf(-c_LAMB[l] * d);
    }
  }
  __syncthreads();
  float a0 = 0.f, a1 = 0.f, a2 = 0.f, a3 = 0.f;
  for (int i = 0; i < deg; ++i) {
    int t = s_tgt[i];
    float val;
    if (f < DIM)        val = memory[(size_t)t * DIM + f] * s_dec[i][f >> 6];
    else if (f < RAW)   val = nodef[(size_t)t * FNODE + (f - DIM)];
    else if (f == RAW)  val = 1.0f;
    else                val = 0.0f;
    a0 += s_w[i][0] * val; a1 += s_w[i][1] * val;
    a2 += s_w[i][2] * val; a3 += s_w[i][3] * val;
  }
  if (f < RAW) {
    size_t b = (size_t)(n * 4) * RAW + f;
    core[b] = a0; core[b + RAW] = a1; core[b + 2 * RAW] = a2; core[b + 3 * RAW] = a3;
  } else if (f == RAW) {
    cnt4[n * 4 + 0] = a0; cnt4[n * 4 + 1] = a1; cnt4[n * 4 + 2] = a2; cnt4[n * 4 + 3] = a3;
  }
}

// ---------------- generic f32 WMMA GEMM ----------------
// C[M,N] = act(alpha * (A @ B) + bias). TA/TB: compile-time transpose flags.
// One 16x32 tile per wave32 (two f32 accumulators sharing the A fragment),
// K unrolled x2 (8 per iteration) + remainder-4 tail. K must be multiple of 4,
// ldA/ldB must be even (true for every call site).
// Out-of-range rows/cols are handled by CLAMPING load addresses (no divergence);
// the clamped lanes compute values that are simply never stored.
// V_WMMA_F32_16X16X4_F32 VGPR layouts:
//   A 16x4: lane<16 -> {K=k0,k0+1}, lane>=16 -> {K=k0+2,k0+3}, M = lane&15
//   B 4x16: same K split, N = lane&15
//   C 16x16: vgpr r -> row r (lane<16) / r+8 (lane>=16), col = lane&15
__device__ __forceinline__ float apply_act(float v, int act) {
  if (act == 1) return fmaxf(v, 0.f);
  if (act == 2) return (v >= 0.f) ? v : 0.01f * v;
  return v;
}

template <int TA, int TB>
__global__ __launch_bounds__(256)
void k_gemm_wmma(const float* __restrict__ A, const float* __restrict__ B,
                 const float* __restrict__ bias, float* __restrict__ C,
                 int M, int N, int K, int ldA, int ldB, int ldC,
                 float alpha, int act) {
  int tilesN2 = (N + 31) >> 5;
  int tilesM  = (M + 15) >> 4;
  int wave = blockIdx.x * (blockDim.x >> 5) + (threadIdx.x >> 5);
  if (wave >= tilesM * tilesN2) return;
  int tm = wave / tilesN2, tn = wave % tilesN2;
  int lane = threadIdx.x & 31;
  int half = lane >> 4;
  int l16  = lane & 15;

  int arow   = tm * 16 + l16;
  int arow_c = (arow < M) ? arow : (M - 1);
  int bcol0  = tn * 32 + l16;
  int bcol1  = bcol0 + 16;
  int bc0    = (bcol0 < N) ? bcol0 : (N - 1);
  int bc1    = (bcol1 < N) ? bcol1 : (N - 1);

  const float* Abase  = TA ? (A + arow_c) : (A + (size_t)arow_c * ldA);
  const float* Bbase0 = TB ? (B + (size_t)bc0 * ldB) : (B + bc0);
  const float* Bbase1 = TB ? (B + (size_t)bc1 * ldB) : (B + bc1);

  v8f acc0 = {0.f, 0.f, 0.f, 0.f, 0.f, 0.f, 0.f, 0.f};
  v8f acc1 = {0.f, 0.f, 0.f, 0.f, 0.f, 0.f, 0.f, 0.f};

  int k0 = 0;
  for (; k0 + 8 <= K; k0 += 8) {
    int ka = k0 + 2 * half;
    v2f a0, a1, b00, b01, b10, b11;
    if (TA) {
      a0.x = Abase[(size_t)ka * ldA];       a0.y = Abase[(size_t)(ka + 1) * ldA];
      a1.x = Abase[(size_t)(ka + 4) * ldA]; a1.y = Abase[(size_t)(ka + 5) * ldA];
    } else {
      a0 = *(const v2f*)(Abase + ka);
      a1 = *(const v2f*)(Abase + ka + 4);
    }
    if (TB) {
      b00 = *(const v2f*)(Bbase0 + ka);
      b01 = *(const v2f*)(Bbase0 + ka + 4);
      b10 = *(const v2f*)(Bbase1 + ka);
      b11 = *(const v2f*)(Bbase1 + ka + 4);
    } else {
      b00.x = Bbase0[(size_t)ka * ldB];       b00.y = Bbase0[(size_t)(ka + 1) * ldB];
      b01.x = Bbase0[(size_t)(ka + 4) * ldB]; b01.y = Bbase0[(size_t)(ka + 5) * ldB];
      b10.x = Bbase1[(size_t)ka * ldB];       b10.y = Bbase1[(size_t)(ka + 1) * ldB];
      b11.x = Bbase1[(size_t)(ka + 4) * ldB]; b11.y = Bbase1[(size_t)(ka + 5) * ldB];
    }
    acc0 = __builtin_amdgcn_wmma_f32_16x16x4_f32(false, a0, false, b00, (short)0, acc0, false, false);
    acc1 = __builtin_amdgcn_wmma_f32_16x16x4_f32(false, a0, false, b10, (short)0, acc1, false, false);
    acc0 = __builtin_amdgcn_wmma_f32_16x16x4_f32(false, a1, false, b01, (short)0, acc0, false, false);
    acc1 = __builtin_amdgcn_wmma_f32_16x16x4_f32(false, a1, false, b11, (short)0, acc1, false, false);
  }
  if (k0 < K) {  // remainder of 4 (K % 8 == 4)
    int ka = k0 + 2 * half;
    v2f a0, b00, b10;
    if (TA) {
      a0.x = Abase[(size_t)ka * ldA]; a0.y = Abase[(size_t)(ka + 1) * ldA];
    } else {
      a0 = *(const v2f*)(Abase + ka);
    }
    if (TB) {
      b00 = *(const v2f*)(Bbase0 + ka);
      b10 = *(const v2f*)(Bbase1 + ka);
    } else {
      b00.x = Bbase0[(size_t)ka * ldB]; b00.y = Bbase0[(size_t)(ka + 1) * ldB];
      b10.x = Bbase1[(size_t)ka * ldB]; b10.y = Bbase1[(size_t)(ka + 1) * ldB];
    }
    acc0 = __builtin_amdgcn_wmma_f32_16x16x4_f32(false, a0, false, b00, (short)0, acc0, false, false);
    acc1 = __builtin_amdgcn_wmma_f32_16x16x4_f32(false, a0, false, b10, (short)0, acc1, false, false);
  }

  // epilogue
  if (bcol0 < N) {
    float bv = bias ? bias[bcol0] : 0.0f;
    #pragma unroll
    for (int r = 0; r < 8; ++r) {
      int row = tm * 16 + r + 8 * half;
      if (row < M) C[(size_t)row * ldC + bcol0] = apply_act(alpha * acc0[r] + bv, act);
    }
  }
  if (bcol1 < N) {
    float bv = bias ? bias[bcol1] : 0.0f;
    #pragma unroll
    for (int r = 0; r < 8; ++r) {
      int row = tm * 16 + r + 8 * half;
      if (row < M) C[(size_t)row * ldC + bcol1] = apply_act(alpha * acc1[r] + bv, act);
    }
  }
}

// ---------------- small helper kernels ----------------
__global__ void k_zero_pad_H(float* H, int rows) {
  int i = blockIdx.x * blockDim.x + threadIdx.x;
  if (i >= rows) return;
  H[(size_t)i * HIDP + 178] = 0.f;
  H[(size_t)i * HIDP + 179] = 0.f;
}

__global__ void k_prep_w2p(const float* __restrict__ W2, float* __restrict__ W2p) {
  int i = blockIdx.x * blockDim.x + threadIdx.x;
  if (i >= HIDP * MSGD) return;
  int k = i >> 6, c = i & 63;
  W2p[i] = (k < HID) ? W2[k * MSGD + c] : 0.f;
}

__global__ void k_row_div(float* __restrict__ v, const float* __restrict__ cnt4, long total) {
  long i = (long)blockIdx.x * blockDim.x + threadIdx.x;
  if (i >= total) return;
  v[i] = v[i] / cnt4[i >> 6];
}

__global__ void k_static(const float* __restrict__ memory, const float* __restrict__ lu,
                         const float* __restrict__ begin_p, float* __restrict__ out, long total) {
  long i = (long)blockIdx.x * blockDim.x + threadIdx.x;
  if (i >= total) return;
  int n = (int)(i >> 8), f = (int)(i & 255), l = f >> 6;
  out[i] = memory[i] * expf(-c_LAMB[l] * (begin_p[0] - lu[n]));
}

// region q,k,v: one block (256) per region, x row cached in LDS
__global__ void k_region_qkv(const float* __restrict__ rmem, const float* __restrict__ Wq,
                             const float* __restrict__ Wk, const float* __restrict__ Wv,
                             float* __restrict__ q, float* __restrict__ k, float* __restrict__ v) {
  int r = blockIdx.x, t = threadIdx.x;
  __shared__ float x[DIM];
  x[t] = rmem[(size_t)r * DIM + t];
  __syncthreads();
  if (t < 64) {
    float aq = 0.f, ak = 0.f;
    for (int i = 0; i < DIM; ++i) { aq += x[i] * Wq[i * 64 + t]; ak += x[i] * Wk[i * 64 + t]; }
    q[r * 64 + t] = aq; k[r * 64 + t] = ak;
  }
  float av = 0.f;
  for (int i = 0; i < DIM; ++i) av += x[i] * Wv[i * DIM + t];
  v[(size_t)r * DIM + t] = av;
}

// region attention: scores, top-3 mask from adj_sub, masked softmax, y = x + attn@v
__global__ void k_region_attn(const float* __restrict__ q, const float* __restrict__ k,
                              const float* __restrict__ v, const float* __restrict__ rmem,
                              const float* __restrict__ adj, const int* __restrict__ ridx,
                              float* __restrict__ y) {
  int r = blockIdx.x, t = threadIdx.x;
  __shared__ float qrow[64];
  __shared__ float s[NREG];
  __shared__ float a[NREG];
  __shared__ int   top[3];
  __shared__ float attnv[3];
  if (t < 64) qrow[t] = q[r * 64 + t];
  __syncthreads();
  if (t < NREG) {
    float d = 0.f;
    for (int i = 0; i < 64; ++i) d += qrow[i] * k[t * 64 + i];
    s[t] = d * 0.125f;                         // / sqrt(64)
    a[t] = adj[ridx[r] * NREG + ridx[t]];      // adj_sub row r
  }
  __syncthreads();
  if (t == 0) {
    for (int j = 0; j < 3; ++j) {              // top-3, earliest index on ties
      int bi = 0; float bv = -1e30f;
      for (int i = 0; i < NREG; ++i) if (a[i] > bv) { bv = a[i]; bi = i; }
      top[j] = bi; a[bi] = -1e30f;
    }
    float m = s[top[0]];
    if (s[top[1]] > m) m = s[top[1]];
    if (s[top[2]] > m) m = s[top[2]];
    float e0 = expf(s[top[0]] - m), e1 = expf(s[top[1]] - m), e2 = expf(s[top[2]] - m);
    float inv = 1.0f / (e0 + e1 + e2);         // masked entries underflow to exactly 0
    attnv[0] = e0 * inv; attnv[1] = e1 * inv; attnv[2] = e2 * inv;
  }
  __syncthreads();
  float acc = rmem[(size_t)r * DIM + t];
  #pragma unroll
  for (int j = 0; j < 3; ++j) acc += attnv[j] * v[(size_t)top[j] * DIM + t];
  y[(size_t)r * DIM + t] = acc;
}

// region_emb = y + relu(y@W1+b1)@W2 + b2
__global__ void k_region_mlp(const float* __restrict__ y, const float* __restrict__ W1,
                             const float* __restrict__ b1, const float* __restrict__ W2,
                             const float* __restrict__ b2, float* __restrict__ remb) {
  int r = blockIdx.x, t = threadIdx.x;
  __shared__ float yy[DIM];
  __shared__ float h[128];
  yy[t] = y[(size_t)r * DIM + t];
  __syncthreads();
  if (t < 128) {
    float d = b1[t];
    for (int i = 0; i < DIM; ++i) d += yy[i] * W1[i * 128 + t];
    h[t] = fmaxf(d, 0.f);
  }
  __syncthreads();
  float d = b2[t];
  for (int i = 0; i < 128; ++i) d += h[i] * W2[i * DIM + t];
  remb[(size_t)r * DIM + t] = yy[t] + d;
}

// softmax over rows (axis=1): one block per row
__global__ void k_softmax_rows(const float* __restrict__ in, float* __restrict__ out,
                               int rows, int cols) {
  int r = blockIdx.x, t = threadIdx.x;
  __shared__ float red[256];
  float m = -1e30f;
  for (int c = t; c < cols; c += 256) m = fmaxf(m, in[(size_t)r * cols + c]);
  red[t] = m; __syncthreads();
  for (int s2 = 128; s2 > 0; s2 >>= 1) { if (t < s2) red[t] = fmaxf(red[t], red[t + s2]); __syncthreads(); }
  m = red[0]; __syncthreads();
  float sum = 0.f;
  for (int c = t; c < cols; c += 256) sum += expf(in[(size_t)r * cols + c] - m);
  red[t] = sum; __syncthreads();
  for (int s2 = 128; s2 > 0; s2 >>= 1) { if (t < s2) red[t] += red[t + s2]; __syncthreads(); }
  float inv = 1.0f / red[0];
  for (int c = t; c < cols; c += 256) out[(size_t)r * cols + c] = expf(in[(size_t)r * cols + c] - m) * inv;
}

// softmax over columns (axis=0): one thread per column
__global__ void k_softmax_cols(const float* __restrict__ in, float* __restrict__ out,
                               int rows, int cols) {
  int c = blockIdx.x * blockDim.x + threadIdx.x;
  if (c >= cols) return;
  float m = -1e30f;
  for (int r = 0; r < rows; ++r) m = fmaxf(m, in[(size_t)r * cols + c]);
  float sum = 0.f;
  for (int r = 0; r < rows; ++r) sum += expf(in[(size_t)r * cols + c] - m);
  float inv = 1.0f / sum;
  for (int r = 0; r < rows; ++r) out[(size_t)r * cols + c] = expf(in[(size_t)r * cols + c] - m) * inv;
}

// recent_node = (memory*exp(-L*(t_last-lu)) + msg) * exp(-L*(now-t_last))
__global__ void k_node_update(const float* __restrict__ memory, const float* __restrict__ lu,
                              const float* __restrict__ tlast, const float* __restrict__ msg,
                              const float* __restrict__ now_p, float* __restrict__ recent, long total) {
  long i = (long)blockIdx.x * blockDim.x + threadIdx.x;
  if (i >= total) return;
  int n = (int)(i >> 8), f = (int)(i & 255), l = f >> 6;
  float tl = tlast[n];
  float upd = memory[i] * expf(-c_LAMB[l] * (tl - lu[n])) + msg[i];   // msg is (n*4+l)*64+m == i
  recent[i] = upd * expf(-c_LAMB[l] * (now_p[0] - tl));
}

// recent = mem*exp(-L*(now-lu)) + msgs   (regions / graphs)
__global__ void k_decay_add(const float* __restrict__ mem, const float* __restrict__ lu,
                            const float* __restrict__ msgs, const float* __restrict__ now_p,
                            float* __restrict__ out, long total) {
  long i = (long)blockIdx.x * blockDim.x + threadIdx.x;
  if (i >= total) return;
  int r = (int)(i >> 8), f = (int)(i & 255), l = f >> 6;
  out[i] = mem[i] * expf(-c_LAMB[l] * (now_p[0] - lu[r])) + msgs[i];
}

// gn_small[r,:] = sum_g g2r[g,r] * recent_graph[g,:]
__global__ void k_gn_small(const float* __restrict__ g2r, const float* __restrict__ rg,
                           float* __restrict__ out) {
  int r = blockIdx.x, t = threadIdx.x;
  float acc = 0.f;
  #pragma unroll
  for (int g = 0; g < NGRAPH; ++g) acc += g2r[g * NREG + r] * rg[g * DIM + t];
  out[(size_t)r * DIM + t] = acc;
}

__global__ void k_final(const float* __restrict__ semb, const float* __restrict__ spatial,
                        const float* __restrict__ lamb_p, float* __restrict__ out, long total) {
  long i = (long)blockIdx.x * blockDim.x + threadIdx.x;
  if (i >= total) return;
  float la = lamb_p[0];
  out[i] = la * semb[i] + (1.0f - la) * spatial[i];
}

// ---------------- host launcher ----------------
extern "C" void kernel_launch(void* const* d_in, const int* in_sizes, int n_in,
                              void* d_out, int out_size, void* d_ws, size_t ws_size,
                              hipStream_t stream) {
  // inputs in setup_inputs() dict order (params expanded in place)
  const float* memory         = (const float*)d_in[0];
  const float* last_update    = (const float*)d_in[1];
  const float* region_memory  = (const float*)d_in[2];
  const float* region_lu      = (const float*)d_in[3];
  const float* graph_memory   = (const float*)d_in[4];
  const float* graph_lu       = (const float*)d_in[5];
  const float* node_features  = (const float*)d_in[6];
  const float* edge_timediff  = (const float*)d_in[7];
  const float* timestamps     = (const float*)d_in[8];
  const float* now_time       = (const float*)d_in[9];
  const float* begin_time     = (const float*)d_in[10];
  const float* p_static_emb   = (const float*)d_in[11];
  const float* p_adj          = (const float*)d_in[12];
  const float* p_msg_W1       = (const float*)d_in[13];
  const float* p_msg_b1       = (const float*)d_in[14];
  const float* p_msg_W2       = (const float*)d_in[15];
  const float* p_msg_b2       = (const float*)d_in[16];
  const float* p_Qr_W         = (const float*)d_in[17];
  const float* p_Kr_W         = (const float*)d_in[18];
  const float* p_Vr_W         = (const float*)d_in[19];
  const float* p_Qg_W         = (const float*)d_in[20];
  const float* p_Kg_W         = (const float*)d_in[21];
  const float* p_Vg_W         = (const float*)d_in[22];
  const float* p_ffr_W1       = (const float*)d_in[23];
  const float* p_ffr_b1       = (const float*)d_in[24];
  const float* p_ffr_W2       = (const float*)d_in[25];
  const float* p_ffr_b2       = (const float*)d_in[26];
  const float* p_ffg_W1       = (const float*)d_in[27];
  const float* p_ffg_b1       = (const float*)d_in[28];
  const float* p_ffg_W2       = (const float*)d_in[29];
  const float* p_ffg_b2       = (const float*)d_in[30];
  const float* p_et_W         = (const float*)d_in[31];
  const float* p_et_b         = (const float*)d_in[32];
  const float* p_st_W         = (const float*)d_in[33];
  const float* p_st_b         = (const float*)d_in[34];
  const float* p_lamb         = (const float*)d_in[35];
  const float* p_r_Wq         = (const float*)d_in[36];
  const float* p_r_Wk         = (const float*)d_in[37];
  const float* p_r_Wv         = (const float*)d_in[38];
  const float* p_r_mW1        = (const float*)d_in[39];
  const float* p_r_mb1        = (const float*)d_in[40];
  const float* p_r_mW2        = (const float*)d_in[41];
  const float* p_r_mb2        = (const float*)d_in[42];
  const int*   source_nodes   = (const int*)d_in[43];
  const int*   target_nodes   = (const int*)d_in[44];
  const int*   region_indices = (const int*)d_in[45];
  (void)in_sizes; (void)n_in; (void)out_size; (void)ws_size;

  // ---- workspace bump allocator ----
  size_t off = 0;
  char* base = (char*)d_ws;
  auto alloc_f = [&](size_t n) -> float* {
    float* p = (float*)(base + off);
    off += ((n * sizeof(float)) + 255) & ~(size_t)255;
    return p;
  };
  auto alloc_i = [&](size_t n) -> int* {
    int* p = (int*)(base + off);
    off += ((n * sizeof(int)) + 255) & ~(size_t)255;
    return p;
  };

  float* f_tlast  = alloc_f(NNODE);
  int*   i_cnt    = alloc_i(NNODE);
  int*   i_bucket = alloc_i((size_t)NNODE * MAXDEG);
  float* f_core   = alloc_f((size_t)NNODE * 4 * RAW);   // (40000,356)
  float* f_cnt4   = alloc_f((size_t)NNODE * 4);
  float* f_H      = alloc_f((size_t)NNODE * 4 * HIDP);  // (40000,180)
  float* f_msg    = alloc_f((size_t)NNODE * 4 * MSGD);  // (40000,64)
  float* f_vmsg   = alloc_f((size_t)NNODE * 4 * MSGD);  // (40000,64) == (10000,256)
  float* f_W2p    = alloc_f((size_t)HIDP * MSGD);
  float* f_static = alloc_f((size_t)NNODE * DIM);
  float* f_Kr     = alloc_f((size_t)NNODE * 64);
  float* f_q      = alloc_f((size_t)NREG * 64);
  float* f_k      = alloc_f((size_t)NREG * 64);
  float* f_v      = alloc_f((size_t)NREG * DIM);
  float* f_y      = alloc_f((size_t)NREG * DIM);
  float* f_remb   = alloc_f((size_t)NREG * DIM);
  float* f_qr     = alloc_f((size_t)NREG * 64);
  float* f_Ar     = alloc_f((size_t)NREG * NNODE);
  float* f_sar    = alloc_f((size_t)NREG * NNODE);
  float* f_r2n    = alloc_f((size_t)NREG * NNODE);
  float* f_rmid   = alloc_f((size_t)NREG * DIM);        // (400,64)
  float* f_t1     = alloc_f((size_t)NREG * 4 * 64);
  float* f_rmsg   = alloc_f((size_t)NREG * 4 * 64);
  float* f_rv     = alloc_f((size_t)NREG * 4 * 64);
  float* f_qg     = alloc_f((size_t)NGRAPH * 64);
  float* f_Kgr    = alloc_f((size_t)NREG * 64);
  float* f_Ag     = alloc_f((size_t)NGRAPH * NREG);
  float* f_sag    = alloc_f((size_t)NGRAPH * NREG);
  float* f_g2r    = alloc_f((size_t)NGRAPH * NREG);
  float* f_gmid   = alloc_f((size_t)NGRAPH * DIM);      // (16,64)
  float* f_t2     = alloc_f((size_t)16 * 64);
  float* f_gmsg   = alloc_f((size_t)16 * 64);
  float* f_rnode  = alloc_f((size_t)NNODE * DIM);       // recent_node
  float* f_rreg   = alloc_f((size_t)NREG * DIM);        // recent_region
  float* f_rgra   = alloc_f((size_t)NGRAPH * DIM);      // recent_graph
  float* f_gns    = alloc_f((size_t)NREG * DIM);        // g2r.T @ recent_graph
  float* f_regn   = alloc_f((size_t)NNODE * DIM);       // region_node
  float* f_gran   = alloc_f((size_t)NNODE * DIM);       // graph_node
  float* f_et     = alloc_f((size_t)NNODE * 192);       // et concat (N,192)
  float* f_spat   = alloc_f((size_t)NNODE * EMB);

  auto gemm = [&](const float* A, const float* B, const float* bias, float* C,
                  int M, int N, int K, int ldA, int ldB, int ldC,
                  int tA, int tB, float alpha, int act) {
    int tiles = ((M + 15) / 16) * ((N + 31) / 32);
    int blocks = (tiles + 7) / 8;
    if (tA == 0 && tB == 0)
      k_gemm_wmma<0, 0><<<blocks, 256, 0, stream>>>(A, B, bias, C, M, N, K, ldA, ldB, ldC, alpha, act);
    else if (tA == 0)
      k_gemm_wmma<0, 1><<<blocks, 256, 0, stream>>>(A, B, bias, C, M, N, K, ldA, ldB, ldC, alpha, act);
    else
      k_gemm_wmma<1, 0><<<blocks, 256, 0, stream>>>(A, B, bias, C, M, N, K, ldA, ldB, ldC, alpha, act);
  };

  // -------- edge phase --------
  k_zero32<<<(NNODE + 255) / 256, 256, 0, stream>>>((unsigned int*)f_tlast, NNODE);
  k_zero32<<<(NNODE + 255) / 256, 256, 0, stream>>>((unsigned int*)i_cnt, NNODE);
  k_edge_scan<<<(NEDGE + 255) / 256, 256, 0, stream>>>(source_nodes, timestamps,
      (unsigned int*)f_tlast, i_cnt, i_bucket, NEDGE);
  k_node_agg<<<NNODE, 384, 0, stream>>>(memory, node_features, timestamps, edge_timediff,
      target_nodes, f_tlast, i_cnt, i_bucket, f_core, f_cnt4);

  // -------- msg MLP (WMMA) --------
  gemm(f_core, p_msg_W1, p_msg_b1, f_H, NNODE * 4, HID, RAW, RAW, HID, HIDP, 0, 0, 1.0f, 1);
  k_zero_pad_H<<<(NNODE * 4 + 255) / 256, 256, 0, stream>>>(f_H, NNODE * 4);
  k_prep_w2p<<<(HIDP * MSGD + 255) / 256, 256, 0, stream>>>(p_msg_W2, f_W2p);
  gemm(f_H, f_W2p, p_msg_b2, f_msg, NNODE * 4, MSGD, HIDP, HIDP, MSGD, MSGD, 0, 0, 1.0f, 0);

  // -------- v_msgs = (core @ Vr_W) / cnt --------
  gemm(f_core, p_Vr_W, nullptr, f_vmsg, NNODE * 4, MSGD, RAW, RAW, MSGD, MSGD, 0, 0, 1.0f, 0);
  k_row_div<<<(NNODE * 4 * MSGD + 255) / 256, 256, 0, stream>>>(f_vmsg, f_cnt4,
      (long)NNODE * 4 * MSGD);

  // -------- static node embedding + Kr --------
  k_static<<<((long)NNODE * DIM + 255) / 256, 256, 0, stream>>>(memory, last_update,
      begin_time, f_static, (long)NNODE * DIM);
  gemm(f_static, p_Kr_W, nullptr, f_Kr, NNODE, 64, DIM, DIM, 64, 64, 0, 0, 1.0f, 0);

  // -------- region self-attention --------
  k_region_qkv<<<NREG, 256, 0, stream>>>(region_memory, p_r_Wq, p_r_Wk, p_r_Wv, f_q, f_k, f_v);
  k_region_attn<<<NREG, 256, 0, stream>>>(f_q, f_k, f_v, region_memory, p_adj,
      region_indices, f_y);
  k_region_mlp<<<NREG, 256, 0, stream>>>(f_y, p_r_mW1, p_r_mb1, p_r_mW2, p_r_mb2, f_remb);

  // -------- A_r, softmaxes --------
  gemm(f_remb, p_Qr_W, nullptr, f_qr, NREG, 64, DIM, DIM, 64, 64, 0, 0, 1.0f, 0);
  gemm(f_qr, f_Kr, nullptr, f_Ar, NREG, NNODE, 64, 64, 64, NNODE, 0, 1, 0.125f, 0);
  k_softmax_rows<<<NREG, 256, 0, stream>>>(f_Ar, f_sar, NREG, NNODE);
  k_softmax_cols<<<(NNODE + 255) / 256, 256, 0, stream>>>(f_Ar, f_r2n, NREG, NNODE);

  // -------- region_mid, region messages --------
  gemm(f_sar, f_vmsg, nullptr, f_rmid, NREG, DIM, NNODE, NNODE, DIM, DIM, 0, 0, 1.0f, 0);
  gemm(f_rmid, p_ffr_W1, p_ffr_b1, f_t1, NREG * 4, 64, 64, 64, 64, 64, 0, 0, 1.0f, 2);
  gemm(f_t1, p_ffr_W2, p_ffr_b2, f_rmsg, NREG * 4, 64, 64, 64, 64, 64, 0, 0, 1.0f, 2);

  // -------- graph path --------
  gemm(graph_memory, p_Qg_W, nullptr, f_qg, NGRAPH, 64, DIM, DIM, 64, 64, 0, 0, 1.0f, 0);
  gemm(f_remb, p_Kg_W, nullptr, f_Kgr, NREG, 64, DIM, DIM, 64, 64, 0, 0, 1.0f, 0);
  gemm(f_qg, f_Kgr, nullptr, f_Ag, NGRAPH, NREG, 64, 64, 64, NREG, 0, 1, 0.125f, 0);
  k_softmax_rows<<<NGRAPH, 256, 0, stream>>>(f_Ag, f_sag, NGRAPH, NREG);
  k_softmax_cols<<<1, 256, 0, stream>>>(f_Ag, f_g2r, NGRAPH, NREG);
  gemm(f_rmid, p_Vg_W, nullptr, f_rv, NREG * 4, 64, 64, 64, 64, 64, 0, 0, 1.0f, 0);
  gemm(f_sag, f_rv, nullptr, f_gmid, NGRAPH, DIM, NREG, NREG, DIM, DIM, 0, 0, 1.0f, 0);
  gemm(f_gmid, p_ffg_W1, p_ffg_b1, f_t2, 16, 64, 64, 64, 64, 64, 0, 0, 1.0f, 2);
  gemm(f_t2, p_ffg_W2, p_ffg_b2, f_gmsg, 16, 64, 64, 64, 64, 64, 0, 0, 1.0f, 2);

  // -------- memory updates --------
  k_node_update<<<((long)NNODE * DIM + 255) / 256, 256, 0, stream>>>(memory, last_update,
      f_tlast, f_msg, now_time, f_rnode, (long)NNODE * DIM);
  k_decay_add<<<((long)NREG * DIM + 255) / 256, 256, 0, stream>>>(region_memory, region_lu,
      f_rmsg, now_time, f_rreg, (long)NREG * DIM);
  k_decay_add<<<((long)NGRAPH * DIM + 255) / 256, 256, 0, stream>>>(graph_memory, graph_lu,
      f_gmsg, now_time, f_rgra, (long)NGRAPH * DIM);

  // -------- region_node / graph_node --------
  gemm(f_r2n, f_rreg, nullptr, f_regn, NNODE, DIM, NREG, NNODE, DIM, DIM, 1, 0, 1.0f, 0);
  k_gn_small<<<NREG, 256, 0, stream>>>(f_g2r, f_rgra, f_gns);
  gemm(f_r2n, f_gns, nullptr, f_gran, NNODE, DIM, NREG, NNODE, DIM, DIM, 1, 0, 1.0f, 0);

  // -------- et (3 parts, interleaved into (N,192)) --------
  gemm(f_rnode, p_et_W, p_et_b, f_et + 0,   NNODE, 64, DIM, DIM, 64, 192, 0, 0, 1.0f, 2);
  gemm(f_regn,  p_et_W, p_et_b, f_et + 64,  NNODE, 64, DIM, DIM, 64, 192, 0, 0, 1.0f, 2);
  gemm(f_gran,  p_et_W, p_et_b, f_et + 128, NNODE, 64, DIM, DIM, 64, 192, 0, 0, 1.0f, 2);

  // -------- spatial + final blend --------
  gemm(f_et, p_st_W, p_st_b, f_spat, NNODE, EMB, 192, 192, EMB, EMB, 0, 0, 1.0f, 2);
  k_final<<<((long)NNODE * EMB + 255) / 256, 256, 0, stream>>>(p_static_emb, f_spat,
      p_lamb, (float*)d_out, (long)NNODE * EMB);
}